// GnnModel_58325655879922
// MI455X (gfx1250) — compile-verified
//
#include <hip/hip_runtime.h>
#include <hip/hip_bf16.h>
#include <math.h>

// ---------------------------------------------------------------------------
// GCN on MI455X (gfx1250): fp32 WMMA for the dense GEMMs, L2-resident gather
// + f32 atomic scatter for the edge aggregation (the memory-bound part).
// ---------------------------------------------------------------------------

typedef float v2f __attribute__((ext_vector_type(2)));
typedef float v8f __attribute__((ext_vector_type(8)));

#define DFEAT 128

// ---------------------------------------------------------------------------
// GEMM: C[N,128] = A[N,128] @ W[128,128], fp32, via V_WMMA_F32_16X16X4_F32.
// Block = 256 threads = 8 wave32s. Each wave owns a 16x16 output tile:
// rows [16*blockIdx.x, +16), cols [16*wave, +16). K loop = 32 WMMA steps.
//
// A-frag layout (ISA 7.12.2, 32-bit A 16x4): lanes 0-15 -> M=lane, K={k0,k0+1};
// lanes 16-31 -> M=lane-16, K={k0+2,k0+3}; VGPR0=first K, VGPR1=second.
// B-frag (4x16, rows striped across lanes per VGPR): VGPR r holds
// K = r + 2*(lane>=16), N = lane&15.
// C/D (16x16 f32): VGPR r -> row r + 8*(lane>=16), col lane&15.
// ---------------------------------------------------------------------------
__global__ void __launch_bounds__(256)
gemm_f32_wmma_kernel(const float* __restrict__ A, const float* __restrict__ W,
                     float* __restrict__ C, int nrows) {
  const int lane    = threadIdx.x & 31;
  const int wave    = threadIdx.x >> 5;          // 0..7 -> column tile
  const int m0      = blockIdx.x * 16;
  const int n0      = wave * 16;
  const int laneLo  = lane & 15;
  const int hi      = lane >> 4;                 // 0 or 1
  if (m0 >= nrows) return;

  const float* Arow = A + (size_t)(m0 + laneLo) * DFEAT;
  const float* Wcol = W + n0 + laneLo;

  v8f acc = {};
#pragma unroll
  for (int k0 = 0; k0 < DFEAT; k0 += 4) {
    v2f a, b;
    const int ka = k0 + 2 * hi;
    a.x = Arow[ka];
    a.y = Arow[ka + 1];
    b.x = Wcol[(size_t)(k0 + 2 * hi)     * DFEAT];
    b.y = Wcol[(size_t)(k0 + 2 * hi + 1) * DFEAT];
    // (neg_a, A, neg_b, B, c_mod, C, reuse_a, reuse_b)
    acc = __builtin_amdgcn_wmma_f32_16x16x4_f32(false, a, false, b,
                                                (short)0, acc, false, false);
  }

  float* Cout = C + (size_t)m0 * DFEAT + n0 + laneLo;
#pragma unroll
  for (int r = 0; r < 8; ++r) {
    Cout[(size_t)(r + 8 * hi) * DFEAT] = acc[r];
  }
}

// ---------------------------------------------------------------------------
// Degree / normalization kernels. deg includes the self-loop (+1).
// ---------------------------------------------------------------------------
__global__ void deg_init_kernel(float* __restrict__ deg, int n) {
  int i = blockIdx.x * blockDim.x + threadIdx.x;
  if (i < n) deg[i] = 1.0f;   // self-loop contribution
}

__global__ void deg_count_kernel(const long long* __restrict__ row,
                                 float* __restrict__ deg, int e) {
  int i = blockIdx.x * blockDim.x + threadIdx.x;
  if (i < e) atomicAdd(&deg[(int)row[i]], 1.0f);
}

__global__ void dinv_kernel(float* __restrict__ deg, int n) {
  int i = blockIdx.x * blockDim.x + threadIdx.x;
  if (i < n) deg[i] = rsqrtf(deg[i]);   // deg >= 1 always
}

// ---------------------------------------------------------------------------
// Self-loop term doubles as output init: out[i,:] = dinv[i]^2 * xw[i,:].
// One wave per node, float4 per lane (32*4 = 128 features).
// ---------------------------------------------------------------------------
__global__ void __launch_bounds__(256)
agg_self_kernel(const float* __restrict__ xw, const float* __restrict__ dinv,
                float* __restrict__ out, int n) {
  int t = blockIdx.x * blockDim.x + threadIdx.x;
  int node = t >> 5;
  if (node >= n) return;
  int lane = t & 31;
  float w = dinv[node] * dinv[node];
  float4 v = ((const float4*)(xw + (size_t)node * DFEAT))[lane];
  float4 o = make_float4(w * v.x, w * v.y, w * v.z, w * v.w);
  ((float4*)(out + (size_t)node * DFEAT))[lane] = o;
}

// ---------------------------------------------------------------------------
// Edge scatter: out[row,:] += dinv[row]*dinv[col] * xw[col,:].
// One wave per edge; gather hits L2 (xw is 25.6 MB << 192 MB L2);
// scatter via global_atomic_add_f32.
// ---------------------------------------------------------------------------
__global__ void __launch_bounds__(256)
agg_edges_kernel(const float* __restrict__ xw, const float* __restrict__ dinv,
                 const long long* __restrict__ row,
                 const long long* __restrict__ col,
                 float* __restrict__ out, int e) {
  int t = blockIdx.x * blockDim.x + threadIdx.x;
  int edge = t >> 5;
  if (edge >= e) return;
  int lane = t & 31;
  int r = (int)row[edge];
  int c = (int)col[edge];
  float norm = dinv[r] * dinv[c];
  float4 v = ((const float4*)(xw + (size_t)c * DFEAT))[lane];
  float* dst = out + (size_t)r * DFEAT + lane * 4;
  atomicAdd(dst + 0, norm * v.x);
  atomicAdd(dst + 1, norm * v.y);
  atomicAdd(dst + 2, norm * v.z);
  atomicAdd(dst + 3, norm * v.w);
}

// ---------------------------------------------------------------------------
// Activation: h = tanh(h) * 5, in place.
// ---------------------------------------------------------------------------
__global__ void act_kernel(float* __restrict__ h, int n4) {
  int i = blockIdx.x * blockDim.x + threadIdx.x;
  if (i >= n4) return;
  float4 v = ((float4*)h)[i];
  v.x = tanhf(v.x) * 5.0f;
  v.y = tanhf(v.y) * 5.0f;
  v.z = tanhf(v.z) * 5.0f;
  v.w = tanhf(v.w) * 5.0f;
  ((float4*)h)[i] = v;
}

static inline int cdiv(long long a, long long b) { return (int)((a + b - 1) / b); }

extern "C" void kernel_launch(void* const* d_in, const int* in_sizes, int n_in,
                              void* d_out, int out_size, void* d_ws, size_t ws_size,
                              hipStream_t stream) {
  // Inputs (setup_inputs order): x [N,128] f32, edge_index [2,E] i64,
  // W1/W2/W3 [128,128] f32. Output: [N,128] f32.
  const float*     x  = (const float*)d_in[0];
  const long long* ei = (const long long*)d_in[1];
  const float*     W1 = (const float*)d_in[2];
  const float*     W2 = (const float*)d_in[3];
  const float*     W3 = (const float*)d_in[4];
  float*           out = (float*)d_out;

  const int N = in_sizes[0] / DFEAT;
  const int E = in_sizes[1] / 2;
  const long long* row = ei;
  const long long* col = ei + E;

  // Workspace: dinv[N] | bufA[N*128] | bufB[N*128]
  char* ws = (char*)d_ws;
  size_t dinv_bytes = (((size_t)N * sizeof(float)) + 255) & ~(size_t)255;
  float* dinv = (float*)ws;
  float* bufA = (float*)(ws + dinv_bytes);
  float* bufB = bufA + (size_t)N * DFEAT;

  const int B = 256;
  const int gN    = cdiv(N, B);            // node-per-thread grids
  const int gE    = cdiv(E, B);            // edge-per-thread grids
  const int gNW   = cdiv((long long)N * 32, B);   // wave-per-node grids
  const int gEW   = cdiv((long long)E * 32, B);   // wave-per-edge grids
  const int gGemm = cdiv(N, 16);           // 16-row strips
  const int gAct  = cdiv((long long)N * DFEAT / 4, B);

  // --- degree / normalization (recomputed each call: deterministic) ---
  deg_init_kernel<<<gN, B, 0, stream>>>(dinv, N);
  deg_count_kernel<<<gE, B, 0, stream>>>(row, dinv, E);
  dinv_kernel<<<gN, B, 0, stream>>>(dinv, N);

  // --- layer 1: x @ W1 -> bufA; aggregate -> bufB; tanh*5 ---
  gemm_f32_wmma_kernel<<<gGemm, B, 0, stream>>>(x, W1, bufA, N);
  agg_self_kernel<<<gNW, B, 0, stream>>>(bufA, dinv, bufB, N);
  agg_edges_kernel<<<gEW, B, 0, stream>>>(bufA, dinv, row, col, bufB, E);
  act_kernel<<<gAct, B, 0, stream>>>(bufB, (int)((long long)N * DFEAT / 4));

  // --- layer 2: h1 @ W2 -> bufA; aggregate -> bufB; tanh*5 ---
  gemm_f32_wmma_kernel<<<gGemm, B, 0, stream>>>(bufB, W2, bufA, N);
  agg_self_kernel<<<gNW, B, 0, stream>>>(bufA, dinv, bufB, N);
  agg_edges_kernel<<<gEW, B, 0, stream>>>(bufA, dinv, row, col, bufB, E);
  act_kernel<<<gAct, B, 0, stream>>>(bufB, (int)((long long)N * DFEAT / 4));

  // --- layer 3: h2 @ W3 -> bufA; aggregate -> d_out (no activation) ---
  gemm_f32_wmma_kernel<<<gGemm, B, 0, stream>>>(bufB, W3, bufA, N);
  agg_self_kernel<<<gNW, B, 0, stream>>>(bufA, dinv, out, N);
  agg_edges_kernel<<<gEW, B, 0, stream>>>(bufA, dinv, row, col, out, E);
}